// GNNmodel_2929167695879
// MI455X (gfx1250) — compile-verified
//
#include <hip/hip_runtime.h>

typedef __attribute__((ext_vector_type(2))) float v2f;
typedef __attribute__((ext_vector_type(8))) float v8f;

#define HID 16

// ---------------- zero accumulators ----------------
__global__ void k_zero(float* __restrict__ p, long long n) {
  long long i = (long long)blockIdx.x * blockDim.x + threadIdx.x;
  if (i < n) p[i] = 0.0f;
}

// ---------------- layer 1 scatter: agg1[dst] += x[src] ----------------
__global__ void k_scatter1(const float* __restrict__ x,
                           const int* __restrict__ src,
                           const int* __restrict__ dst,
                           float* __restrict__ agg1, int E) {
  int e = blockIdx.x * blockDim.x + threadIdx.x;
  if (e >= E) return;
  int s = src[e], d = dst[e];
  atomicAdd(&agg1[d], x[s]);
}

// ---------------- encoder: z1[i][h] = relu(agg1[i]*wr[h] + x[i]*wo[h] + b[h]) ----------------
__global__ void k_encode(const float* __restrict__ x, const float* __restrict__ agg1,
                         const float* __restrict__ w_rel, const float* __restrict__ w_root,
                         const float* __restrict__ b, float* __restrict__ z1, int N) {
  int t = blockIdx.x * blockDim.x + threadIdx.x;
  int i = t >> 4, h = t & 15;
  if (i >= N) return;
  float v = agg1[i] * w_rel[h] + x[i] * w_root[h] + b[h];
  z1[t] = v > 0.0f ? v : 0.0f;
}

// ---------------- 16-wide scatter: agg[dst*16+h] += z[src*16+h] ----------------
// lane = (edge, h): 16 consecutive lanes share an edge -> coalesced 64B gather,
// 16 consecutive f32 atomics per edge (no return -> STOREcnt path).
__global__ void k_scatter16(const float* __restrict__ z,
                            const int* __restrict__ src,
                            const int* __restrict__ dst,
                            float* __restrict__ agg, int E) {
  long long t = (long long)blockIdx.x * blockDim.x + threadIdx.x;
  int e = (int)(t >> 4);
  int h = (int)(t & 15);
  if (e >= E) return;
  int s = src[e], d = dst[e];
  atomicAdd(&agg[(long long)d * HID + h], z[(long long)s * HID + h]);
}

// ---------------- processor layer via V_WMMA_F32_16X16X4_F32 ----------------
// One wave32 per 16-node tile. z2 = [agg2 | z1] (16x32) @ [Wrel ; Wroot] (32x16) + b
// done as 8 K=4 f32 WMMA steps (exact f32, no precision loss). Bias is folded
// into the initial accumulator (D column of every acc element is lane&15).
__global__ void k_proc_wmma(const float* __restrict__ agg2, const float* __restrict__ z1,
                            const float* __restrict__ w_rel, const float* __restrict__ w_root,
                            const float* __restrict__ b, float* __restrict__ z2, int N) {
  int lane = threadIdx.x & 31;
  int wave = (blockIdx.x * (blockDim.x >> 5)) + (threadIdx.x >> 5);
  int row0 = wave * 16;                  // first node of this tile (wave-uniform)
  if (row0 >= N) return;                 // uniform exit: EXEC stays all-1s otherwise

  int m = lane & 15;                     // A: row index / B,D: column index
  int khalf = (lane >> 4) << 1;          // lanes 16-31 hold K+2,K+3 of each chunk
  int node = row0 + m;
  if (node >= N) node = N - 1;           // clamp loads; keeps EXEC full

  const float* aRow = agg2 + (long long)node * HID;
  const float* zRow = z1   + (long long)node * HID;

  float bias = b[m];
  v8f acc;
#pragma unroll
  for (int v = 0; v < 8; ++v) acc[v] = bias;   // bias pre-seeded into C

#pragma unroll
  for (int c = 0; c < 4; ++c) {          // K = 0..15  : agg2 @ w_rel
    int k = c * 4 + khalf;
    v2f a, wb;
    a.x  = aRow[k];               a.y  = aRow[k + 1];
    wb.x = w_rel[k * HID + m];    wb.y = w_rel[(k + 1) * HID + m];
    acc = __builtin_amdgcn_wmma_f32_16x16x4_f32(false, a, false, wb,
                                                (short)0, acc, false, false);
  }
#pragma unroll
  for (int c = 0; c < 4; ++c) {          // K = 16..31 : z1 @ w_root
    int k = c * 4 + khalf;
    v2f a, wb;
    a.x  = zRow[k];               a.y  = zRow[k + 1];
    wb.x = w_root[k * HID + m];   wb.y = w_root[(k + 1) * HID + m];
    acc = __builtin_amdgcn_wmma_f32_16x16x4_f32(false, a, false, wb,
                                                (short)0, acc, false, false);
  }

  // D layout: VGPR v -> row (v + 8*(lane>=16)), col = lane&15
  int mbase = (lane >> 4) * 8;
  if (row0 + 16 <= N) {
    // wave-uniform fast path: stores with immediate offsets, no exec juggling
    float* out = z2 + (long long)(row0 + mbase) * HID + m;
#pragma unroll
    for (int v = 0; v < 8; ++v) out[v * HID] = acc[v];
  } else {
#pragma unroll
    for (int v = 0; v < 8; ++v) {
      int r = row0 + mbase + v;
      if (r < N) z2[(long long)r * HID + m] = acc[v];
    }
  }
}

// ---------------- decoder: y[i] = relu(dot(agg3_i, wr) + dot(z2_i, wo) + b) ----------------
__global__ void k_decode(const float* __restrict__ agg3, const float* __restrict__ z2,
                         const float* __restrict__ w_rel, const float* __restrict__ w_root,
                         const float* __restrict__ b, float* __restrict__ y, int N) {
  int i = blockIdx.x * blockDim.x + threadIdx.x;
  if (i >= N) return;
  const float* a = agg3 + (long long)i * HID;
  const float* z = z2   + (long long)i * HID;
  float s = b[0];
#pragma unroll
  for (int k = 0; k < HID; ++k) s += a[k] * w_rel[k] + z[k] * w_root[k];
  y[i] = s > 0.0f ? s : 0.0f;
}

extern "C" void kernel_launch(void* const* d_in, const int* in_sizes, int n_in,
                              void* d_out, int out_size, void* d_ws, size_t ws_size,
                              hipStream_t stream) {
  const float* x          = (const float*)d_in[0];
  const int*   edge       = (const int*)d_in[1];
  const float* enc_w_rel  = (const float*)d_in[2];
  const float* enc_w_root = (const float*)d_in[3];
  const float* enc_b      = (const float*)d_in[4];
  const float* proc_w_rel = (const float*)d_in[5];
  const float* proc_w_root= (const float*)d_in[6];
  const float* proc_b     = (const float*)d_in[7];
  const float* dec_w_rel  = (const float*)d_in[8];
  const float* dec_w_root = (const float*)d_in[9];
  const float* dec_b      = (const float*)d_in[10];

  const int N = in_sizes[0];        // 100000 nodes
  const int E = in_sizes[1] / 2;    // 3.2M edges
  const int* src = edge;            // edge_index row 0
  const int* dst = edge + E;        // edge_index row 1

  // workspace layout (floats): accumulators first so one zero pass covers them
  float* ws   = (float*)d_ws;
  float* agg1 = ws;                               // N
  float* agg2 = agg1 + (size_t)N;                 // 16N
  float* agg3 = agg2 + (size_t)16 * N;            // 16N
  float* z1   = agg3 + (size_t)16 * N;            // 16N
  float* z2   = z1   + (size_t)16 * N;            // 16N

  const int TPB = 256;
  long long nzero = (long long)N * 33;            // agg1+agg2+agg3
  k_zero<<<(unsigned)((nzero + TPB - 1) / TPB), TPB, 0, stream>>>(agg1, nzero);

  // layer 1
  k_scatter1<<<(E + TPB - 1) / TPB, TPB, 0, stream>>>(x, src, dst, agg1, E);
  k_encode<<<(unsigned)(((long long)N * 16 + TPB - 1) / TPB), TPB, 0, stream>>>(
      x, agg1, enc_w_rel, enc_w_root, enc_b, z1, N);

  // layer 2
  long long t16 = (long long)E * 16;
  k_scatter16<<<(unsigned)((t16 + TPB - 1) / TPB), TPB, 0, stream>>>(z1, src, dst, agg2, E);
  int tiles = (N + 15) / 16;                      // one wave32 per tile
  int waves_per_blk = TPB / 32;
  k_proc_wmma<<<(tiles + waves_per_blk - 1) / waves_per_blk, TPB, 0, stream>>>(
      agg2, z1, proc_w_rel, proc_w_root, proc_b, z2, N);

  // layer 3
  k_scatter16<<<(unsigned)((t16 + TPB - 1) / TPB), TPB, 0, stream>>>(z2, src, dst, agg3, E);
  k_decode<<<(N + TPB - 1) / TPB, TPB, 0, stream>>>(
      agg3, z2, dec_w_rel, dec_w_root, dec_b, (float*)d_out, N);
}